// ECGPrivacyProcessor_89258010345779
// MI455X (gfx1250) — compile-verified
//
#include <hip/hip_runtime.h>
#include <math.h>

#define NTH   512
#define LSIG  16384
#define MHALF 8192
#define PI_F  3.14159265358979323846f

typedef float v2f __attribute__((ext_vector_type(2)));
typedef float v8f __attribute__((ext_vector_type(8)));

__device__ __forceinline__ float2 mkf2(float a, float b){ float2 r; r.x=a; r.y=b; return r; }
__device__ __forceinline__ float2 cmul(float2 a, float2 b){ return mkf2(a.x*b.x - a.y*b.y, a.x*b.y + a.y*b.x); }
__device__ __forceinline__ float2 cadd(float2 a, float2 b){ return mkf2(a.x+b.x, a.y+b.y); }
__device__ __forceinline__ float2 csub(float2 a, float2 b){ return mkf2(a.x-b.x, a.y-b.y); }
__device__ __forceinline__ float2 cscl(float2 a, float s){ return mkf2(a.x*s, a.y*s); }
__device__ __forceinline__ float2 cconj(float2 a){ return mkf2(a.x, -a.y); }

// Stockham radix-2 autosort FFT, N=8192, ping-pong x<->y. Returns buffer holding result.
// sign = -1 forward, +1 inverse (unnormalized).
__device__ float2* fft8192(float2* x, float2* y, int tid, float sign) {
  int n = MHALF, s = 1, ls = 0;
  while (n >= 2) {
    int m = n >> 1;
    float theta = sign * (2.0f * PI_F) / (float)n;
    for (int t = tid; t < (MHALF / 2); t += NTH) {
      int p = t >> ls;
      int q = t & (s - 1);
      float2 a = x[q + s * p];
      float2 b = x[q + s * (p + m)];
      float2 su = cadd(a, b);
      float2 di = csub(a, b);
      float sn, cs;
      __sincosf(theta * (float)p, &sn, &cs);
      y[q + s * (2 * p)]     = su;
      y[q + s * (2 * p + 1)] = cmul(di, mkf2(cs, sn));
    }
    __syncthreads();
    float2* tmp = x; x = y; y = tmp;
    n >>= 1; s <<= 1; ++ls;
  }
  return x;
}

// X[idx] (rfft bin of length-2M real signal) from Z = FFT_M(x_even + i*x_odd).
__device__ __forceinline__ float2 Xat(const float2* Z, int idx) {
  float2 Zk = Z[idx & (MHALF - 1)];
  float2 Zm = cconj(Z[(MHALF - idx) & (MHALF - 1)]);
  float2 su = cscl(cadd(Zk, Zm), 0.5f);
  float2 di = cscl(csub(Zk, Zm), 0.5f);
  float ang = -PI_F * (float)idx * (1.0f / (float)MHALF);
  float sn, cs; __sincosf(ang, &sn, &cs);
  float2 t = cmul(mkf2(cs, sn), di);
  return mkf2(su.x + t.y, su.y - t.x);   // su + (-i)*t
}

// lin_interp of time-domain phase (len 16384) at rfft bin kidx in [0,8192]
__device__ __forceinline__ float phinterp(const float* PH, int kidx) {
  float src = ((float)kidx + 0.5f) * ((float)LSIG / 8193.0f) - 0.5f;
  src = fminf(fmaxf(src, 0.0f), (float)(LSIG - 1));
  int i0 = (int)src;
  int i1 = min(i0 + 1, LSIG - 1);
  float w = src - (float)i0;
  return PH[i0] * (1.0f - w) + PH[i1] * w;
}

__global__ __launch_bounds__(NTH, 1)
void ecg_fused_kernel(const float* __restrict__ x,
                      const float* __restrict__ w1, const float* __restrict__ b1,
                      const float* __restrict__ w2, const float* __restrict__ b2,
                      const float* __restrict__ w3, const float* __restrict__ b3,
                      const float* __restrict__ w4, const float* __restrict__ b4,
                      const float* __restrict__ amp_s, const float* __restrict__ phase_s,
                      const float* __restrict__ morph_s,
                      float* __restrict__ out)
{
  // ---- LDS layout (~235 KB of the 320 KB WGP budget) ----
  __shared__ __align__(16) float XA[LSIG];     // x; later FFT ping-pong buffer B
  __shared__ __align__(16) float PH[LSIG];     // per-sample phase from CNN
  __shared__ __align__(16) float CBf[LSIG];    // cumsum; later FFT ping-pong buffer A
  __shared__ float w2r[32 * 48];               // conv2 weights, K = tap*16 + c
  __shared__ float w3r[16 * 32];               // conv3 weights
  __shared__ float w1s[48], b1s[16], b2s[32], b3s[16], w4s[48], b4s[4];
  __shared__ float scanp[NTH];
  __shared__ __align__(16) float WSCR[16 * 512]; // per-wave conv scratch (512 f32 each)

  const int tid  = threadIdx.x;
  const int bb   = blockIdx.x;
  const size_t base = (size_t)bb * LSIG;

  // ================= Phase 0: stage signal + weights into LDS =================
  {
    const float4* x4 = (const float4*)(x + base);
    float4* XA4 = (float4*)XA;
    __builtin_prefetch(x4 + tid, 0, 0);        // global_prefetch_b8
    for (int i = tid; i < LSIG / 4; i += NTH) XA4[i] = x4[i];

    for (int i = tid; i < 32 * 48; i += NTH) {
      int o = i / 48, k = i % 48;
      int tap = k >> 4, c = k & 15;
      w2r[i] = w2[o * 48 + c * 3 + tap];
    }
    for (int i = tid; i < 16 * 32; i += NTH) w3r[i] = w3[i];
    if (tid < 48) w1s[tid] = w1[tid];
    if (tid < 16) b1s[tid] = b1[tid];
    if (tid < 32) b2s[tid] = b2[tid];
    if (tid < 16) b3s[tid] = b3[tid];
    if (tid < 48) w4s[tid] = w4[tid];
    if (tid < 3)  b4s[tid] = b4[tid];
  }
  __syncthreads();

  // ================= Phase 1: peak/valley indicator + block-wide cumsum =======
  {
    int base0 = tid * 32;
    float run = 0.0f;
    for (int i = 0; i < 32; ++i) {
      int t = base0 + i;
      float ind = 0.0f;
      if (t >= 1 && t <= LSIG - 2) {
        float xm = XA[t], a = XA[t - 1], b = XA[t + 1];
        ind = (xm > a && xm > b) ? 1.0f : ((xm < a && xm < b) ? -1.0f : 0.0f);
      }
      run += ind;
      CBf[t] = run;          // local inclusive prefix
    }
    scanp[tid] = run;
  }
  __syncthreads();
  if (tid < 32) {            // wave 0: exclusive scan of 512 partials
    float tmp[16];
    float s = 0.0f;
    #pragma unroll
    for (int i = 0; i < 16; ++i) { tmp[i] = s; s += scanp[tid * 16 + i]; }
    float inc = s;
    #pragma unroll
    for (int d = 1; d < 32; d <<= 1) {
      float o = __shfl_up(inc, d, 32);
      if (tid >= d) inc += o;
    }
    float excl = inc - s;
    #pragma unroll
    for (int i = 0; i < 16; ++i) scanp[tid * 16 + i] = excl + tmp[i];
  }
  __syncthreads();
  {
    float off = scanp[tid];
    int base0 = tid * 32;
    for (int i = 0; i < 32; ++i) CBf[base0 + i] += off;   // CBf[t] = csum[t]
  }
  __syncthreads();

  // ================= Phase 2: conv stack via V_WMMA_F32_16X16X4_F32 ===========
  {
    const int wid  = tid >> 5;
    const int lane = tid & 31;
    const int half = lane >> 4;     // K-half of the wave per WMMA A/B layout
    const int l15  = lane & 15;     // M (A) / N (B) index
    float* wscr = &WSCR[wid << 9];

    // Preload A operands (weights) in WMMA register layout, once per wave.
    v2f a2lo[12], a2hi[12], a3c[8];
    #pragma unroll
    for (int kk = 0; kk < 12; ++kk) {
      int kA = 4 * kk + 2 * half;
      a2lo[kk] = (v2f){ w2r[l15 * 48 + kA],        w2r[l15 * 48 + kA + 1] };
      a2hi[kk] = (v2f){ w2r[(l15 + 16) * 48 + kA], w2r[(l15 + 16) * 48 + kA + 1] };
    }
    #pragma unroll
    for (int kk = 0; kk < 8; ++kk) {
      int kA = 4 * kk + 2 * half;
      a3c[kk] = (v2f){ w3r[l15 * 32 + kA], w3r[l15 * 32 + kA + 1] };
    }
    const float ampS = amp_s[0];
    const float phS  = phase_s[0] * PI_F;
    const float moS  = morph_s[0];

    for (int ch = wid; ch < LSIG / 16; ch += 16) {   // uniform trip count per wave
      const int t0   = ch << 4;
      const int posC = t0 + l15;
      float xw[5];
      #pragma unroll
      for (int i = 0; i < 5; ++i) {
        int ix = posC - 2 + i;
        xw[i] = ((unsigned)ix < (unsigned)LSIG) ? XA[ix] : 0.0f;  // conv1 zero-pad
      }

      // conv2 = GEMM (32 x 48) x (48 x 16): 12 K-steps, two 16x16 output tiles
      v8f acc0 = {}, acc1 = {};
      #pragma unroll
      for (int kk = 0; kk < 12; ++kk) {
        int tap = kk >> 2;
        int c0  = ((kk & 3) << 2) + 2 * half;
        int tpos = posC - 1 + tap;
        bool inb = (unsigned)tpos < (unsigned)LSIG;  // conv2 zero-pad of f1
        float xm1 = xw[tap], x0v = xw[tap + 1], xp1 = xw[tap + 2];
        float f0 = fmaxf(0.0f, b1s[c0] +
                   w1s[c0 * 3] * xm1 + w1s[c0 * 3 + 1] * x0v + w1s[c0 * 3 + 2] * xp1);
        float f1 = fmaxf(0.0f, b1s[c0 + 1] +
                   w1s[c0 * 3 + 3] * xm1 + w1s[c0 * 3 + 4] * x0v + w1s[c0 * 3 + 5] * xp1);
        v2f bvec = { inb ? f0 : 0.0f, inb ? f1 : 0.0f };
        acc0 = __builtin_amdgcn_wmma_f32_16x16x4_f32(false, a2lo[kk], false, bvec,
                                                     (short)0, acc0, false, false);
        acc1 = __builtin_amdgcn_wmma_f32_16x16x4_f32(false, a2hi[kk], false, bvec,
                                                     (short)0, acc1, false, false);
      }

      // f2 = relu(acc + b2)  -> wscr[n][o] (n-major, 32 floats per n)
      #pragma unroll
      for (int v = 0; v < 8; ++v) {
        int oA = v + 8 * half;
        wscr[l15 * 32 + oA]      = fmaxf(0.0f, acc0[v] + b2s[oA]);
        wscr[l15 * 32 + 16 + oA] = fmaxf(0.0f, acc1[v] + b2s[16 + oA]);
      }
      __builtin_amdgcn_wave_barrier();

      // conv3 = GEMM (16 x 32) x (32 x 16): 8 K-steps
      v8f acc2 = {};
      #pragma unroll
      for (int kk = 0; kk < 8; ++kk) {
        int kA = 4 * kk + 2 * half;
        v2f bv = { wscr[l15 * 32 + kA], wscr[l15 * 32 + kA + 1] };
        acc2 = __builtin_amdgcn_wmma_f32_16x16x4_f32(false, a3c[kk], false, bv,
                                                     (short)0, acc2, false, false);
      }
      __builtin_amdgcn_wave_barrier();
      // p3 = relu(acc2 + b3) -> wscr[0..255] as [n][m] (f2 region, already consumed)
      #pragma unroll
      for (int v = 0; v < 8; ++v) {
        int m = v + 8 * half;
        wscr[l15 * 16 + m] = fmaxf(0.0f, acc2[v] + b3s[m]);
      }
      __builtin_amdgcn_wave_barrier();

      // conv4 (3 x 16) + tanh; half0 computes j=0 and j=2, half1 computes j=1
      {
        int j1 = half;
        float sA = b4s[j1], sB = b4s[2];
        #pragma unroll
        for (int c = 0; c < 16; ++c) {
          float pv = wscr[l15 * 16 + c];
          sA += w4s[j1 * 16 + c] * pv;
          sB += w4s[32 + c] * pv;
        }
        wscr[256 + j1 * 16 + l15] = tanhf(sA);
        if (half == 0) wscr[256 + 32 + l15] = tanhf(sB);
      }
      __builtin_amdgcn_wave_barrier();

      // amp/phase/morph + morphology mask + partial output
      if (lane < 16) {
        int t = t0 + lane;
        float amp = wscr[256 + lane]      * ampS;
        float phv = wscr[256 + 16 + lane] * phS;
        float mor = wscr[256 + 32 + lane] * moS;
        PH[t] = phv;
        float chi = CBf[min(t + 10, LSIG - 1)];
        float clo = (t >= 10) ? CBf[t - 10] : 0.0f;
        out[base + t] = XA[t] * amp + mor * (chi - clo);   // x*amp + morph*mask
      }
      __builtin_amdgcn_wave_barrier();
    }
  }
  __threadfence();        // partial outputs reach L2 before final atomic adds
  __syncthreads();

  // ================= Phase 3: pack real signal for half-length complex FFT ====
  float2* CBc = (float2*)CBf;
  float2* XAc = (float2*)XA;
  for (int n = tid; n < MHALF; n += NTH)
    CBc[n] = mkf2(XA[2 * n], XA[2 * n + 1]);
  __syncthreads();

  // ================= Phase 4: forward FFT =====================================
  float2* zp = fft8192(CBc, XAc, tid, -1.0f);   // Z in zp
  float2* wp = (zp == XAc) ? CBc : XAc;

  // ========= Phase 5: unpack rfft + phase rotation + repack for irfft =========
  for (int k = tid; k < MHALF; k += NTH) {
    float2 Xk = Xat(zp, k);
    float2 Xm = Xat(zp, MHALF - k);
    float p0 = phinterp(PH, k);
    float p1 = phinterp(PH, MHALF - k);
    float s0, c0, s1, c1;
    __sincosf(p0, &s0, &c0);
    __sincosf(p1, &s1, &c1);
    float2 Yk  = cmul(Xk, mkf2(c0, s0));
    float2 Ymc = cconj(cmul(Xm, mkf2(c1, s1)));
    float2 su = cscl(cadd(Yk, Ymc), 0.5f);
    float2 di = cscl(csub(Yk, Ymc), 0.5f);
    float ang = PI_F * (float)k * (1.0f / (float)MHALF);
    float sn, cs; __sincosf(ang, &sn, &cs);
    float2 t = cmul(mkf2(cs, sn), di);
    wp[k] = mkf2(su.x - t.y, su.y + t.x);       // su + i*t
  }
  __syncthreads();

  // ================= Phase 6: inverse FFT =====================================
  float2* zf = fft8192(wp, zp, tid, +1.0f);

  // ================= Phase 7: out += irfft result =============================
  const float invM = 1.0f / (float)MHALF;
  for (int n = tid; n < MHALF; n += NTH) {
    float2 z = zf[n];
    atomicAdd(&out[base + 2 * n],     z.x * invM);
    atomicAdd(&out[base + 2 * n + 1], z.y * invM);
  }
}

extern "C" void kernel_launch(void* const* d_in, const int* in_sizes, int n_in,
                              void* d_out, int out_size, void* d_ws, size_t ws_size,
                              hipStream_t stream) {
  const float* x       = (const float*)d_in[0];
  const float* w1      = (const float*)d_in[1];
  const float* b1      = (const float*)d_in[2];
  const float* w2      = (const float*)d_in[3];
  const float* b2      = (const float*)d_in[4];
  const float* w3      = (const float*)d_in[5];
  const float* b3      = (const float*)d_in[6];
  const float* w4      = (const float*)d_in[7];
  const float* b4      = (const float*)d_in[8];
  const float* amp_s   = (const float*)d_in[9];
  const float* phase_s = (const float*)d_in[10];
  const float* morph_s = (const float*)d_in[11];
  float* out = (float*)d_out;

  int Bn = in_sizes[0] / LSIG;
  if (Bn < 1) Bn = 1;
  ecg_fused_kernel<<<dim3(Bn), dim3(NTH), 0, stream>>>(
      x, w1, b1, w2, b2, w3, b3, w4, b4, amp_s, phase_s, morph_s, out);
}